// LoRACausalSelfAttention_64441689309651
// MI455X (gfx1250) — compile-verified
//
#include <hip/hip_runtime.h>

#define H_DIM 1024
#define T_SEQ 2048
#define B_SZ 4
#define N_HEADS 16
#define HEAD_DIM 64
#define RANK 8
#define LORA_SCALE 2.0f

typedef __attribute__((ext_vector_type(2))) float v2f;
typedef __attribute__((ext_vector_type(8))) float v8f;
typedef int v4i_vs __attribute__((vector_size(16)));  // matches builtin param

// ---------------------------------------------------------------------------
// CDNA5 async global->LDS copy (ASYNCcnt-tracked, bypasses VGPRs).
// Signature (probe-confirmed): (v4i AS1* src, v4i AS3* dst, imm off, imm cpol)
// ---------------------------------------------------------------------------
#if __has_builtin(__builtin_amdgcn_global_load_async_to_lds_b128)
#define HAS_ASYNC_LDS 1
#else
#define HAS_ASYNC_LDS 0
#endif

__device__ __forceinline__ void async_copy16(const float* gsrc, float* ldst) {
#if HAS_ASYNC_LDS
  auto g = (__attribute__((address_space(1))) v4i_vs*)(uintptr_t)gsrc;
  auto l = (__attribute__((address_space(3))) v4i_vs*)(uint32_t)(uintptr_t)ldst;
  __builtin_amdgcn_global_load_async_to_lds_b128(g, l, 0, 0);
#else
  *(float4*)ldst = *(const float4*)gsrc;
#endif
}

__device__ __forceinline__ void async_wait_all() {
#if HAS_ASYNC_LDS
#if __has_builtin(__builtin_amdgcn_s_wait_asynccnt)
  __builtin_amdgcn_s_wait_asynccnt(0);
#else
  asm volatile("s_wait_asynccnt 0x0" ::: "memory");
#endif
#endif
}

// ----------------------------------------------------------------------------
// Kernel 1: W_eff[o][h] = W[o][h] + LORA_SCALE * sum_r B[o][r] * A[r][h]
// ----------------------------------------------------------------------------
__global__ __launch_bounds__(256) void build_weff_kernel(
    const float* __restrict__ W, const float* __restrict__ A,
    const float* __restrict__ Bm, float* __restrict__ Weff) {
  const int idx = blockIdx.x * 256 + threadIdx.x;  // over H*H
  const int o = idx >> 10;
  const int h = idx & (H_DIM - 1);
  float s = 0.f;
#pragma unroll
  for (int r = 0; r < RANK; ++r) s += Bm[o * RANK + r] * A[r * H_DIM + h];
  Weff[idx] = W[idx] + LORA_SCALE * s;
}

// ----------------------------------------------------------------------------
// Kernel 2: fused QKV GEMM.  P = X @ Weff^T + bias, scattered to [B,NH,T,HD].
// Block tile 128x128, 4 waves each 64x64 (16 f32 WMMA accumulators).
// Double-buffered async global->LDS staging, KT=16, LDS stride 20 floats
// (conflict-free ds_load_b64 fragment fetches, 16B-aligned rows).
// ----------------------------------------------------------------------------
#define KT 16
#define LDT 20

__device__ __forceinline__ void gemm_stage_tiles(const float* __restrict__ X,
                                                 const float* __restrict__ Wp,
                                                 int m0, int n0, int k0,
                                                 float* As, float* Bs,
                                                 int tid) {
  const int rr = tid >> 2;        // 0..31
  const int cc = (tid & 3) * 4;   // 0..12 step 4
#pragma unroll
  for (int r = 0; r < 4; ++r) {
    const int row = r * 32 + rr;
    async_copy16(&X[(size_t)(m0 + row) * H_DIM + k0 + cc], &As[row * LDT + cc]);
    async_copy16(&Wp[(size_t)(n0 + row) * H_DIM + k0 + cc], &Bs[row * LDT + cc]);
  }
}

__global__ __launch_bounds__(128) void qkv_gemm_kernel(
    const float* __restrict__ X, const float* __restrict__ Weff,
    const float* __restrict__ bq, const float* __restrict__ bk,
    const float* __restrict__ bv, float* __restrict__ qkv) {
  __shared__ float As[2][128 * LDT];
  __shared__ float Bs[2][128 * LDT];

  const int p = blockIdx.z;
  const float* Wp = Weff + (size_t)p * H_DIM * H_DIM;
  const float* bias = (p == 0) ? bq : (p == 1 ? bk : bv);
  float* outp = qkv + (size_t)p * (B_SZ * T_SEQ * H_DIM);

  const int tid = threadIdx.x;
  const int lane = tid & 31;
  const int wave = tid >> 5;
  const int laneL = lane & 15;
  const int hi = lane >> 4;  // 0: K+0/K+1 half, 1: K+2/K+3 half
  const int wm = wave >> 1;
  const int wn = wave & 1;
  const int m0 = blockIdx.y * 128;
  const int n0 = blockIdx.x * 128;

  v8f acc[4][4];
#pragma unroll
  for (int i = 0; i < 4; ++i)
#pragma unroll
    for (int j = 0; j < 4; ++j)
#pragma unroll
      for (int e = 0; e < 8; ++e) acc[i][j][e] = 0.f;

  // prologue: stage first K-tile
  gemm_stage_tiles(X, Wp, m0, n0, 0, As[0], Bs[0], tid);

  int st = 0;
  for (int k0 = 0; k0 < H_DIM; k0 += KT, st ^= 1) {
    async_wait_all();     // this wave's outstanding tile loads done
    __syncthreads();      // -> every wave's loads done; prev compute finished

    if (k0 + KT < H_DIM)  // issue next tile into the other buffer
      gemm_stage_tiles(X, Wp, m0, n0, k0 + KT, As[st ^ 1], Bs[st ^ 1], tid);

    const float* Ac = As[st];
    const float* Bc = Bs[st];
#pragma unroll
    for (int kk = 0; kk < KT / 4; ++kk) {
      const int kcol = kk * 4 + hi * 2;
      v2f af[4], bf[4];
#pragma unroll
      for (int mi = 0; mi < 4; ++mi)
        af[mi] = *(const v2f*)&Ac[(wm * 64 + mi * 16 + laneL) * LDT + kcol];
#pragma unroll
      for (int ni = 0; ni < 4; ++ni)
        bf[ni] = *(const v2f*)&Bc[(wn * 64 + ni * 16 + laneL) * LDT + kcol];
#pragma unroll
      for (int mi = 0; mi < 4; ++mi)
#pragma unroll
        for (int ni = 0; ni < 4; ++ni)
          acc[mi][ni] = __builtin_amdgcn_wmma_f32_16x16x4_f32(
              false, af[mi], false, bf[ni], (short)0, acc[mi][ni], false, false);
    }
  }

  // Epilogue: bias add + scatter to [B, NH, T, HD]
#pragma unroll
  for (int ni = 0; ni < 4; ++ni) {
    const int n = n0 + wn * 64 + ni * 16 + laneL;
    const float bval = bias[n];
    const int head = n >> 6;       // n / HEAD_DIM
    const int d = n & 63;          // n % HEAD_DIM
#pragma unroll
    for (int mi = 0; mi < 4; ++mi) {
#pragma unroll
      for (int vi = 0; vi < 8; ++vi) {
        const int m = m0 + wm * 64 + mi * 16 + vi + hi * 8;
        const int bb = m >> 11;            // m / T_SEQ
        const int t = m & (T_SEQ - 1);
        outp[(((size_t)bb * N_HEADS + head) * T_SEQ + t) * HEAD_DIM + d] =
            acc[mi][ni][vi] + bval;
      }
    }
  }
}

// ----------------------------------------------------------------------------
// Kernel 3: flash attention.  One WG per (b, head, 64-row q block); 4 waves,
// each wave owns 16 q rows.  K/V tiles staged to LDS via async copies (shared
// by all 4 waves); online softmax; P restaged via LDS into WMMA A-layout.
// ----------------------------------------------------------------------------
#define LDK 68  // LDS stride for 64x64 K/V tiles
#define PLD 66  // LDS stride for P staging (16 x 64 tile per wave)

__global__ __launch_bounds__(128) void flash_attn_kernel(
    const float* __restrict__ q, const float* __restrict__ k,
    const float* __restrict__ v, const float* __restrict__ amask,
    float* __restrict__ out) {
  __shared__ float Ks[64 * LDK];
  __shared__ float Vs[64 * LDK];
  __shared__ float Ps[4][16 * PLD];

  const int qb = blockIdx.x;       // q block (64 rows)
  const int h = blockIdx.y;
  const int b = blockIdx.z;
  const int tid = threadIdx.x;
  const int lane = tid & 31;
  const int wave = tid >> 5;
  const int laneL = lane & 15;
  const int hi = lane >> 4;

  const size_t bh = ((size_t)b * N_HEADS + h) * T_SEQ * HEAD_DIM;
  const float* Q = q + bh;
  const float* K = k + bh;
  const float* V = v + bh;
  const float* am = amask + (size_t)b * T_SEQ;

  const int q0 = qb * 64 + wave * 16;  // this wave's first q row

  // Q fragments in registers for all 16 K-dim chunks (A-layout)
  v2f qf[16];
#pragma unroll
  for (int kc = 0; kc < 16; ++kc)
    qf[kc] = *(const v2f*)&Q[(size_t)(q0 + laneL) * HEAD_DIM + kc * 4 + hi * 2];

  v8f o_acc[4];
  float mrow[8], lrow[8];
#pragma unroll
  for (int vi = 0; vi < 8; ++vi) {
    mrow[vi] = -__builtin_inff();
    lrow[vi] = 0.f;
  }
#pragma unroll
  for (int di = 0; di < 4; ++di)
#pragma unroll
    for (int e = 0; e < 8; ++e) o_acc[di][e] = 0.f;

  const int srow = tid >> 4;        // 0..7
  const int scol = (tid & 15) * 4;  // 0..60 step 4

  for (int kb = 0; kb <= qb; ++kb) {
    const int kn0 = kb * 64;

    // ---- stage K/V tiles (64x64 each) into LDS via async copies ----
#pragma unroll
    for (int j = 0; j < 8; ++j) {
      const int row = j * 8 + srow;
      async_copy16(&K[(size_t)(kn0 + row) * HEAD_DIM + scol],
                   &Ks[row * LDK + scol]);
      async_copy16(&V[(size_t)(kn0 + row) * HEAD_DIM + scol],
                   &Vs[row * LDK + scol]);
    }
    async_wait_all();
    __syncthreads();

    // ---- S = (Q K^T) over 64 keys ----
    v8f s[4];
#pragma unroll
    for (int ni = 0; ni < 4; ++ni)
#pragma unroll
      for (int e = 0; e < 8; ++e) s[ni][e] = 0.f;

#pragma unroll
    for (int kc = 0; kc < 16; ++kc) {
      v2f kf[4];
#pragma unroll
      for (int ni = 0; ni < 4; ++ni)
        kf[ni] = *(const v2f*)&Ks[(ni * 16 + laneL) * LDK + kc * 4 + hi * 2];
#pragma unroll
      for (int ni = 0; ni < 4; ++ni)
        s[ni] = __builtin_amdgcn_wmma_f32_16x16x4_f32(
            false, qf[kc], false, kf[ni], (short)0, s[ni], false, false);
    }

    // ---- scale, attention_mask, causal mask (diagonal block only) ----
    const bool diag = (kb == qb);
#pragma unroll
    for (int ni = 0; ni < 4; ++ni) {
      const int kidx = kn0 + ni * 16 + laneL;
      const float amv = am[kidx];
#pragma unroll
      for (int vi = 0; vi < 8; ++vi) {
        float val = s[ni][vi] * 0.125f + amv;
        if (diag) {
          const int qq = q0 + vi + hi * 8;
          if (kidx > qq) val = -__builtin_inff();
        }
        s[ni][vi] = val;
      }
    }

    // ---- online softmax (row = vi + hi*8, data across 16-lane halves) ----
#pragma unroll
    for (int vi = 0; vi < 8; ++vi) {
      float mv = fmaxf(fmaxf(s[0][vi], s[1][vi]), fmaxf(s[2][vi], s[3][vi]));
#pragma unroll
      for (int off = 8; off >= 1; off >>= 1)
        mv = fmaxf(mv, __shfl_xor(mv, off, 32));
      const float newm = fmaxf(mrow[vi], mv);
      const float corr = __expf(mrow[vi] - newm);
      mrow[vi] = newm;
      float psum = 0.f;
#pragma unroll
      for (int ni = 0; ni < 4; ++ni) {
        const float pv = __expf(s[ni][vi] - newm);
        s[ni][vi] = pv;
        psum += pv;
      }
#pragma unroll
      for (int off = 8; off >= 1; off >>= 1)
        psum += __shfl_xor(psum, off, 32);
      lrow[vi] = lrow[vi] * corr + psum;
#pragma unroll
      for (int di = 0; di < 4; ++di) o_acc[di][vi] *= corr;
    }

    // ---- stage P to LDS (C-layout -> A-layout) then P @ V via WMMA ----
    float* Pw = Ps[wave];
#pragma unroll
    for (int ni = 0; ni < 4; ++ni)
#pragma unroll
      for (int vi = 0; vi < 8; ++vi)
        Pw[(vi + hi * 8) * PLD + ni * 16 + laneL] = s[ni][vi];
    __syncthreads();

#pragma unroll
    for (int kc = 0; kc < 16; ++kc) {
      const v2f pf = *(const v2f*)&Pw[laneL * PLD + kc * 4 + hi * 2];
      const float* vbase = &Vs[(kc * 4 + hi * 2) * LDK];
#pragma unroll
      for (int di = 0; di < 4; ++di) {
        const int dn = di * 16 + laneL;
        v2f vf;
        vf[0] = vbase[dn];        // V row kc*4+hi*2
        vf[1] = vbase[LDK + dn];  // V row kc*4+hi*2+1 (fuses to ds_load_2addr)
        o_acc[di] = __builtin_amdgcn_wmma_f32_16x16x4_f32(
            false, pf, false, vf, (short)0, o_acc[di], false, false);
      }
    }
    __syncthreads();  // all waves done with Ks/Vs/Ps before next stage
  }

  // ---- finalize: divide by row sums, write [B, T, H] ----
#pragma unroll
  for (int vi = 0; vi < 8; ++vi) {
    const float linv = 1.0f / lrow[vi];
    const int tq = qb * 64 + wave * 16 + vi + hi * 8;
#pragma unroll
    for (int di = 0; di < 4; ++di) {
      const int dd = di * 16 + laneL;
      out[((size_t)b * T_SEQ + tq) * H_DIM + h * HEAD_DIM + dd] =
          o_acc[di][vi] * linv;
    }
  }
}

// ----------------------------------------------------------------------------
extern "C" void kernel_launch(void* const* d_in, const int* in_sizes, int n_in,
                              void* d_out, int out_size, void* d_ws,
                              size_t ws_size, hipStream_t stream) {
  const float* hs = (const float*)d_in[0];
  const float* amask = (const float*)d_in[1];
  const float* Wq = (const float*)d_in[2];
  const float* bq = (const float*)d_in[3];
  const float* Aq = (const float*)d_in[4];
  const float* Bq = (const float*)d_in[5];
  const float* Wk = (const float*)d_in[6];
  const float* bk = (const float*)d_in[7];
  const float* Ak = (const float*)d_in[8];
  const float* Bk = (const float*)d_in[9];
  const float* Wv = (const float*)d_in[10];
  const float* bv = (const float*)d_in[11];
  const float* Av = (const float*)d_in[12];
  const float* Bv = (const float*)d_in[13];
  float* out = (float*)d_out;

  // Workspace layout: [3][H][H] W_eff, then q,k,v each [B][NH][T][HD]
  float* ws = (float*)d_ws;
  float* weff = ws;
  float* qbuf = ws + (size_t)3 * H_DIM * H_DIM;
  float* kbuf = qbuf + (size_t)B_SZ * T_SEQ * H_DIM;
  float* vbuf = kbuf + (size_t)B_SZ * T_SEQ * H_DIM;

  const dim3 gW((H_DIM * H_DIM) / 256);
  build_weff_kernel<<<gW, 256, 0, stream>>>(Wq, Aq, Bq, weff);
  build_weff_kernel<<<gW, 256, 0, stream>>>(Wk, Ak, Bk, weff + H_DIM * H_DIM);
  build_weff_kernel<<<gW, 256, 0, stream>>>(Wv, Av, Bv, weff + 2 * H_DIM * H_DIM);

  qkv_gemm_kernel<<<dim3(H_DIM / 128, (B_SZ * T_SEQ) / 128, 3), 128, 0,
                    stream>>>(hs, weff, bq, bk, bv, qbuf);

  flash_attn_kernel<<<dim3(T_SEQ / 64, N_HEADS, B_SZ), 128, 0, stream>>>(
      qbuf, kbuf, vbuf, amask, out);
}